// AConnect_82368882803074
// MI455X (gfx1250) — compile-verified
//
#include <hip/hip_runtime.h>
#include <stdint.h>

// ---------------------------------------------------------------------------
// AConnect training forward:  Z[b,o] = sum_i X[b,i]*W[i,o]*Werr[idx[b],i,o]
//                                      + bias[o]*Berr[idx[b],o]
// B=256, D_IN=D_OUT=512.  Memory-bound (256MB of Werr per call, AI ~0.8 F/B,
// floor ~11us @ 23.3TB/s).  Strategy: stream Werr via async global->LDS
// double buffering (ASYNCcnt), do the i-reduction with V_WMMA_F32_16X16X4_F32
// (A = broadcast X chunk, B = W*E tile) -- redundant across M but correct,
// and compute headroom is >10x the memory floor.
// ---------------------------------------------------------------------------

typedef float v2f __attribute__((ext_vector_type(2)));
typedef float v8f __attribute__((ext_vector_type(8)));

#define BDIM      256
#define D_IN      512
#define D_OUT     512
#define TILE_ROWS 8
#define N_STAGES  (D_IN / TILE_ROWS)                 // 64
#define LDS_STRIDE 520                               // floats; pad 8 -> halves use disjoint banks
#define ASYNC_PER_STAGE ((TILE_ROWS * D_OUT * 4) / (BDIM * 16))  // 4

__device__ __forceinline__ uint32_t lds_off_u32(const void* p) {
  // flat shared-aperture address: low 32 bits are the workgroup-relative LDS offset
  return (uint32_t)(uintptr_t)p;
}

__device__ __forceinline__ void async_b128(uint32_t lds_off, uint32_t goff,
                                           const float* sbase) {
  // GVS mode: mem = SADDR(64b sgpr pair) + VADDR(32b vgpr) ; VDST = LDS offset vgpr
  asm volatile("global_load_async_to_lds_b128 %0, %1, %2"
               :
               : "v"(lds_off), "v"(goff), "s"(sbase)
               : "memory");
}

__global__ __launch_bounds__(BDIM) void aconnect_kernel(
    const float* __restrict__ X, const float* __restrict__ W,
    const float* __restrict__ bias, const float* __restrict__ Werr,
    const float* __restrict__ Berr, const int* __restrict__ idx,
    float* __restrict__ Z) {
  __shared__ __align__(16) float ebuf[2][TILE_ROWS * LDS_STRIDE];
  __shared__ float xs[D_IN];

  const int b    = blockIdx.x;
  const int tid  = threadIdx.x;
  const int lane = tid & 31;
  const int wave = tid >> 5;

  const int ib = __builtin_amdgcn_readfirstlane(idx[b]);
  const float* Ebase = Werr + (size_t)ib * (size_t)(D_IN * D_OUT);

  // stage the X row for this sample into LDS (broadcast source for WMMA A)
  xs[tid]       = X[b * D_IN + tid];
  xs[tid + 256] = X[b * D_IN + tid + 256];

  const uint32_t buf0 = lds_off_u32(&ebuf[0][0]);
  const uint32_t buf1 = lds_off_u32(&ebuf[1][0]);

  auto issue_stage = [&](int s, uint32_t bufoff) {
#pragma unroll
    for (int j = 0; j < ASYNC_PER_STAGE; ++j) {
      uint32_t c    = (uint32_t)(tid + BDIM * j) * 16u;  // byte chunk in tile
      uint32_t row  = c >> 11;                           // / (512*4)
      uint32_t colb = c & 2047u;
      uint32_t lo   = bufoff + row * (LDS_STRIDE * 4u) + colb;
      uint32_t go   = (uint32_t)(s * TILE_ROWS + row) * (D_OUT * 4u) + colb;
      async_b128(lo, go, Ebase);
    }
  };

  // per-wave: 4 o-tiles of 16 columns (8 waves * 64 = 512 outputs)
  const int o0   = wave * 64;
  const int colL = lane & 15;
  const int rsel = (lane >> 4) << 1;  // 0 for lanes 0-15 (K0/K1), 2 for 16-31 (K2/K3)

  v8f acc0 = {}, acc1 = {}, acc2 = {}, acc3 = {};

  issue_stage(0, buf0);

  for (int s = 0; s < N_STAGES; ++s) {
    if (s + 1 < N_STAGES) {
      issue_stage(s + 1, ((s + 1) & 1) ? buf1 : buf0);
      asm volatile("s_wait_asynccnt 4" ::: "memory");  // stage s complete (in-order)
    } else {
      asm volatile("s_wait_asynccnt 0" ::: "memory");
    }
    __syncthreads();  // all waves' stage-s async writes + xs stores visible

    const float* eb = (s & 1) ? &ebuf[1][0] : &ebuf[0][0];

#pragma unroll
    for (int i0 = 0; i0 < TILE_ROWS; i0 += 4) {
      const int rl = i0 + rsel;            // local tile row supplying K0 (or K2)
      const int gr = s * TILE_ROWS + rl;   // global i index
      v2f A;
      A[0] = xs[gr];                       // K0 / K2 (uniform per lane-half)
      A[1] = xs[gr + 1];                   // K1 / K3
      const float* er = eb + rl * LDS_STRIDE;
      const float* wr = W + gr * D_OUT;

      {
        int col = o0 + colL;
        v2f Bm;
        Bm[0] = wr[col] * er[col];
        Bm[1] = wr[D_OUT + col] * er[LDS_STRIDE + col];
        acc0 = __builtin_amdgcn_wmma_f32_16x16x4_f32(false, A, false, Bm,
                                                     (short)0, acc0, false, false);
      }
      {
        int col = o0 + 16 + colL;
        v2f Bm;
        Bm[0] = wr[col] * er[col];
        Bm[1] = wr[D_OUT + col] * er[LDS_STRIDE + col];
        acc1 = __builtin_amdgcn_wmma_f32_16x16x4_f32(false, A, false, Bm,
                                                     (short)0, acc1, false, false);
      }
      {
        int col = o0 + 32 + colL;
        v2f Bm;
        Bm[0] = wr[col] * er[col];
        Bm[1] = wr[D_OUT + col] * er[LDS_STRIDE + col];
        acc2 = __builtin_amdgcn_wmma_f32_16x16x4_f32(false, A, false, Bm,
                                                     (short)0, acc2, false, false);
      }
      {
        int col = o0 + 48 + colL;
        v2f Bm;
        Bm[0] = wr[col] * er[col];
        Bm[1] = wr[D_OUT + col] * er[LDS_STRIDE + col];
        acc3 = __builtin_amdgcn_wmma_f32_16x16x4_f32(false, A, false, Bm,
                                                     (short)0, acc3, false, false);
      }
    }
    __syncthreads();  // buffer reuse fence before next issue
  }

  // D layout: vgpr0 lanes 0-15 hold row M=0, N=0..15 (all M rows identical here)
  if (lane < 16) {
    const int o = o0 + lane;
    float* zr = Z + b * D_OUT;
    zr[o]      = acc0[0] + bias[o]      * Berr[ib * D_OUT + o];
    zr[o + 16] = acc1[0] + bias[o + 16] * Berr[ib * D_OUT + o + 16];
    zr[o + 32] = acc2[0] + bias[o + 32] * Berr[ib * D_OUT + o + 32];
    zr[o + 48] = acc3[0] + bias[o + 48] * Berr[ib * D_OUT + o + 48];
  }
}

extern "C" void kernel_launch(void* const* d_in, const int* in_sizes, int n_in,
                              void* d_out, int out_size, void* d_ws, size_t ws_size,
                              hipStream_t stream) {
  const float* X    = (const float*)d_in[0];  // [256,512]
  const float* W    = (const float*)d_in[1];  // [512,512]
  const float* bias = (const float*)d_in[2];  // [1,512]
  const float* Werr = (const float*)d_in[3];  // [1000,512,512]
  const float* Berr = (const float*)d_in[4];  // [1000,1,512]
  const int*   idx  = (const int*)d_in[5];    // [256]
  float* Z = (float*)d_out;                   // [256,512]

  const int B = in_sizes[0] / D_IN;           // 256
  aconnect_kernel<<<B, BDIM, 0, stream>>>(X, W, bias, Werr, Berr, idx, Z);
}